// SpatioConvLayer_85942295593057
// MI455X (gfx1250) — compile-verified
//
#include <hip/hip_runtime.h>
#include <stdint.h>

typedef float v2f __attribute__((ext_vector_type(2)));
typedef float v8f __attribute__((ext_vector_type(8)));

#define B_    8
#define CIN   64
#define T_    12
#define N_    4096
#define KS_   3
#define COUT  64
#define BN    128     // n-columns per block
#define BK    32      // K-chunk (m) per iteration
#define NCHUNK (N_ / BK)

// ---- LDS layout (float offsets), strides padded for b64/b128 align + banks ----
#define TH_OFF  0                            // thT[k][o][i]  : [3][64][66]
#define TH_STR  66
#define XS_SZ   (CIN * 36)                   // Xs[i][m]      : [64][36] per buffer
#define XS_OFF  (TH_OFF + 3*COUT*TH_STR)     // two buffers
#define XS_STR  36
#define YS_OFF  (XS_OFF + 2*XS_SZ)           // Ys[k][o][m]   : [3][64][36]
#define YS_STR  36
#define GS_SZ   (3 * BN * 36)                // Gs[k][n][m]   : [3][128][36] per buffer
#define GS_OFF  (YS_OFF + 3*COUT*YS_STR)     // two buffers
#define GS_STR  36
#define SMEM_FLOATS (GS_OFF + 2*GS_SZ)       // 51840 floats = 207360 B (< 320KB/WGP)

// CDNA5 async global->LDS copy (ASYNCcnt-tracked), 16B per lane.
__device__ __forceinline__ void async_ld_b128(unsigned lds_byte_addr, const float* gptr) {
    asm volatile("global_load_async_to_lds_b128 %0, %1, off"
                 :: "v"(lds_byte_addr),
                    "v"((unsigned long long)(uintptr_t)gptr)
                 : "memory");
}
__device__ __forceinline__ void wait_async0() {
    asm volatile("s_wait_asynccnt 0" ::: "memory");
}

extern "C" __global__ __launch_bounds__(256, 1)
void stgcn_fused_kernel(const float* __restrict__ x,
                        const float* __restrict__ gbasis,
                        const float* __restrict__ theta,
                        const float* __restrict__ bias,
                        float* __restrict__ out)
{
    extern __shared__ float smem[];

    const int tid  = threadIdx.x;
    const int lane = tid & 31;
    const int w    = tid >> 5;        // wave 0..7
    const int l15  = lane & 15;
    const int lh   = lane >> 4;       // 0 / 1 (lane half)

    const int bt = blockIdx.x;        // fast dim: keeps G n-slice hot in L2
    const int b  = bt / T_;
    const int t  = bt - b * T_;
    const int n0 = blockIdx.y * BN;

    const float* xbt = x + (size_t)b * CIN * (T_ * N_) + (size_t)t * N_;

    // ---- stage theta transposed into LDS: thT[k][o][i] ----
    for (int p = 0; p < 48; ++p) {
        int e   = tid + 256 * p;               // e in [0, 12288)
        int i   = e / (COUT * KS_);
        int rem = e - i * (COUT * KS_);
        int o   = rem / KS_;
        int k   = rem - o * KS_;
        smem[TH_OFF + (k * COUT + o) * TH_STR + i] = theta[e];
    }

    v8f ct[4] = {};                    // 4 C-tiles: o-tiles 0..3, this wave's n-strip
    const int yot = w >> 1;            // Y-tile assignment per wave
    const int ymt = w & 1;

    // ---- async stage of one (X,G) K-chunk into buffer `buf` ----
    auto issue_chunk = [&](int m0, int buf) {
        // X chunk: Xs[buf][i][0..31] = x[b,i,t,m0..m0+31]  (512 float4)
        #pragma unroll
        for (int p = 0; p < 2; ++p) {
            int f = tid + 256 * p;             // 0..511
            int i = f >> 3;
            int c = (f & 7) << 2;
            const float* g = xbt + (size_t)i * (T_ * N_) + m0 + c;
            async_ld_b128((unsigned)(uintptr_t)(smem + XS_OFF + buf * XS_SZ
                                                + i * XS_STR + c), g);
        }
        // G chunk: Gs[buf][k][r][0..31] = G[k][n0+r][m0..m0+31] (3072 float4)
        #pragma unroll
        for (int p = 0; p < 12; ++p) {
            int f = tid + 256 * p;             // 0..3071
            int k = f >> 10;
            int r = (f >> 3) & 127;
            int c = (f & 7) << 2;
            const float* g = gbasis + (size_t)k * N_ * N_ + (size_t)(n0 + r) * N_ + m0 + c;
            async_ld_b128((unsigned)(uintptr_t)(smem + GS_OFF + buf * GS_SZ
                                                + (k * BN + r) * GS_STR + c), g);
            if (m0 + BK < N_) __builtin_prefetch(g + BK, 0, 0);   // HBM->L2 one chunk ahead
        }
    };

    // prologue: chunk 0 into buffer 0
    issue_chunk(0, 0);
    wait_async0();
    __syncthreads();                   // theta + chunk0 visible

    for (int j = 0; j < NCHUNK; ++j) {
        const int cur = j & 1;
        const int m0  = j * BK;

        // overlap: start next chunk's global->LDS while we compute on `cur`
        if (j + 1 < NCHUNK) issue_chunk(m0 + BK, 1 - cur);

        // ---- Y chunk via WMMA: Y_k[o, m] = sum_i thT[k][o][i] * Xs[i][m] ----
        // wave w computes tile (o-tile yot, m-tile ymt) for each k; K=64 -> 16 steps of 4
        const float* Xc = smem + XS_OFF + cur * XS_SZ;
        #pragma unroll
        for (int k = 0; k < KS_; ++k) {
            v8f acc = {};
            #pragma unroll
            for (int is = 0; is < 16; ++is) {
                // A: theta^T  (M=o across lanes 0-15, K=i pair per VGPR/lane-half)
                v2f a = *(const v2f*)(smem + TH_OFF + (k * COUT + yot * 16 + l15) * TH_STR
                                      + is * 4 + lh * 2);
                // B: X (K=i rows, N=m across lanes 0-15)
                int r0 = is * 4 + lh * 2;
                v2f bv;
                bv.x = Xc[(r0    ) * XS_STR + ymt * 16 + l15];
                bv.y = Xc[(r0 + 1) * XS_STR + ymt * 16 + l15];
                acc = __builtin_amdgcn_wmma_f32_16x16x4_f32(false, a, false, bv,
                                                            (short)0, acc, false, false);
            }
            #pragma unroll
            for (int v = 0; v < 8; ++v) {   // C layout: M = v + 8*lh, N = l15
                smem[YS_OFF + (k * COUT + yot * 16 + v + 8 * lh) * YS_STR
                     + ymt * 16 + l15] = acc[v];
            }
        }
        __syncthreads();   // Ys ready

        // ---- main GEMM: ct[ot] += Y_k[o-tile, m] * G_k^T[m, n-strip] ----
        const float* Gc = smem + GS_OFF + cur * GS_SZ;
        #pragma unroll
        for (int k = 0; k < KS_; ++k) {
            #pragma unroll
            for (int ms = 0; ms < BK / 4; ++ms) {
                // B: G^T -> lane = n (row of Gs), K = m pair per VGPR/lane-half
                v2f bv = *(const v2f*)(Gc + (k * BN + w * 16 + l15) * GS_STR
                                       + ms * 4 + lh * 2);
                #pragma unroll
                for (int ot = 0; ot < 4; ++ot) {
                    v2f a = *(const v2f*)(smem + YS_OFF + (k * COUT + ot * 16 + l15) * YS_STR
                                          + ms * 4 + lh * 2);
                    ct[ot] = __builtin_amdgcn_wmma_f32_16x16x4_f32(false, a, false, bv,
                                                                   (short)0, ct[ot],
                                                                   false, false);
                }
            }
        }

        wait_async0();     // next chunk landed in LDS
        __syncthreads();   // everyone done reading `cur` + Ys; next buffer visible
    }

    // ---- epilogue: + bias + identity residual, ReLU, store ----
    const int n = n0 + w * 16 + l15;
    #pragma unroll
    for (int ot = 0; ot < 4; ++ot) {
        #pragma unroll
        for (int v = 0; v < 8; ++v) {
            int o = ot * 16 + v + 8 * lh;
            size_t oidx = (((size_t)b * COUT + o) * T_ + t) * (size_t)N_ + n;
            float val = ct[ot][v] + bias[o] + x[oidx];
            out[oidx] = fmaxf(val, 0.0f);
        }
    }
}

extern "C" void kernel_launch(void* const* d_in, const int* in_sizes, int n_in,
                              void* d_out, int out_size, void* d_ws, size_t ws_size,
                              hipStream_t stream) {
    const float* x     = (const float*)d_in[0];
    const float* gb    = (const float*)d_in[1];
    const float* theta = (const float*)d_in[2];
    const float* bias  = (const float*)d_in[3];
    float* out = (float*)d_out;

    dim3 grid(B_ * T_, N_ / BN, 1);   // 96 x 32 blocks
    dim3 blk(256, 1, 1);              // 8 wave32 waves
    size_t smem_bytes = (size_t)SMEM_FLOATS * sizeof(float);
    hipLaunchKernelGGL(stgcn_fused_kernel, grid, blk, smem_bytes, stream,
                       x, gb, theta, bias, out);
}